// Sender_9603546874479
// MI455X (gfx1250) — compile-verified
//
#include <hip/hip_runtime.h>
#include <cmath>

typedef float v2f __attribute__((ext_vector_type(2)));
typedef float v8f __attribute__((ext_vector_type(8)));

#define BB   1024
#define VV   8192
#define EE   64
#define HH   64
#define LL   16
#define SOS  1
#define EOS  2
#define NSEG 8
#define SEGC (VV / NSEG)   // 1024 columns per segment
#define TPW  16            // 16 col-tiles per wave (4 waves * 16 * 16 = 1024 = SEGC)

// ---- output layout (flat, return order) ----
constexpr size_t OUT_TOK = 0;                                   // [B, L+1, V]
constexpr size_t OUT_SEQ = OUT_TOK + (size_t)BB * (LL + 1) * VV;// [B]
constexpr size_t OUT_ENT = OUT_SEQ + BB;                        // scalar
constexpr size_t OUT_HID = OUT_ENT + 1;                         // [B, H]
constexpr size_t OUT_EMB = OUT_HID + (size_t)BB * HH;           // [B, L, E]

// ---- workspace layout (in floats) ----
constexpr size_t WS_H   = 0;                    // [B*H]
constexpr size_t WS_C   = WS_H + (size_t)BB*HH; // [B*H]
constexpr size_t WS_ENT = WS_C + (size_t)BB*HH; // [B]
constexpr size_t WS_PM  = WS_ENT + BB;          // [B*NSEG]
constexpr size_t WS_PS  = WS_PM + (size_t)BB*NSEG;
constexpr size_t WS_PT  = WS_PS + (size_t)BB*NSEG;
constexpr size_t WS_PBV = WS_PT + (size_t)BB*NSEG;
constexpr size_t WS_TOK = WS_PBV + (size_t)BB*NSEG; // int [B]
constexpr size_t WS_SEQ = WS_TOK + BB;              // int [B]
constexpr size_t WS_PBI = WS_SEQ + BB;              // int [B*NSEG]

__device__ __forceinline__ float sigm(float x) { return 1.0f / (1.0f + __expf(-x)); }

// -------------------- init --------------------
__global__ void k_init(const float* __restrict__ hin, float* __restrict__ h,
                       float* __restrict__ c, float* __restrict__ ent,
                       int* __restrict__ tok, int* __restrict__ seq,
                       float* __restrict__ out) {
  int i = blockIdx.x * blockDim.x + threadIdx.x;
  if (i < BB * HH) {
    float v = hin[i];
    h[i] = v;
    c[i] = 0.0f;
    out[OUT_HID + i] = v;                      // hidden_state passthrough output
  }
  if (i < BB) {
    tok[i] = SOS;
    seq[i] = LL + 1;
    ent[i] = 0.0f;
    out[OUT_TOK + (size_t)i * (LL + 1) * VV + SOS] = 1.0f;   // sos one-hot at t=0
  }
}

// -------------------- per step: gather emb + gates + LSTM cell --------------------
__global__ void k_lstm(const float* __restrict__ emb_tab, const float* __restrict__ Wih,
                       const float* __restrict__ Whh, const float* __restrict__ bih,
                       const float* __restrict__ bhh, const int* __restrict__ tok,
                       float* __restrict__ h, float* __restrict__ c,
                       float* __restrict__ out_embeds, int step) {
  __shared__ float se[EE];
  __shared__ float sh[HH];
  __shared__ float sg[4 * HH];
  const int b = blockIdx.x;
  const int t = threadIdx.x;                   // 0..255
  if (t < EE) {
    float e = emb_tab[(size_t)tok[b] * EE + t];
    se[t] = e;
    out_embeds[((size_t)b * LL + step) * EE + t] = e;
    sh[t] = h[(size_t)b * HH + t];
  }
  __syncthreads();
  float acc = bih[t] + bhh[t];
  const float* wi = Wih + (size_t)t * EE;
  const float* wh = Whh + (size_t)t * HH;
#pragma unroll 8
  for (int k = 0; k < EE; ++k) acc += se[k] * wi[k];
#pragma unroll 8
  for (int k = 0; k < HH; ++k) acc += sh[k] * wh[k];
  sg[t] = acc;
  __syncthreads();
  if (t < HH) {
    float ii = sg[t], ff = sg[HH + t], gg = sg[2 * HH + t], oo = sg[3 * HH + t];
    float cn = sigm(ff) * c[(size_t)b * HH + t] + sigm(ii) * tanhf(gg);
    float hn = sigm(oo) * tanhf(cn);
    c[(size_t)b * HH + t] = cn;
    h[(size_t)b * HH + t] = hn;
  }
}

// -------------------- per step: WMMA logits + fused online softmax stats --------------------
// grid: (B/16, NSEG), block: 128 (4 waves). Each wave: 16 col-tiles of 16 cols, K=64.
__global__ void k_logits(const float* __restrict__ h, const float* __restrict__ Wout,
                         const float* __restrict__ bout, const float* __restrict__ gum,
                         int step, float* __restrict__ pm, float* __restrict__ ps,
                         float* __restrict__ pt, float* __restrict__ pbv,
                         int* __restrict__ pbi) {
  __shared__ float tileS[4][256];
  __shared__ float sm_[128], ss_[128], st_[128], sbv_[128];
  __shared__ int sbi_[128];

  const int row0  = blockIdx.x * 16;
  const int seg   = blockIdx.y;
  const int col00 = seg * SEGC;
  const int tid   = threadIdx.x;
  const int w     = tid >> 5;
  const int lane  = tid & 31;
  const int R     = lane & 15;        // row within tile (A: M, B: N)
  const int half  = lane >> 4;        // lane half selects K pair / column half
  const int koff  = half * 2;

  // A fragments: h rows row0..row0+15, K=0..63 in 16 chunks of 4
  const float* hrow = h + (size_t)(row0 + R) * HH;
  v2f a[16];
#pragma unroll
  for (int kk = 0; kk < 16; ++kk) {
    a[kk].x = hrow[kk * 4 + koff];
    a[kk].y = hrow[kk * 4 + koff + 1];
  }

  float m = -3.0e38f, s = 0.0f, tsum = 0.0f, bv = -3.0e38f;
  int bi = 0;
  const int rbase = (R & 7) * 32 + ((R >> 3) << 4);   // LDS base for row R
  const float* grow = gum + ((size_t)step * BB + row0 + R) * VV;

  for (int t = 0; t < TPW; ++t) {
    const int col0 = col00 + (w * TPW + t) * 16;
    v8f acc = {0.f, 0.f, 0.f, 0.f, 0.f, 0.f, 0.f, 0.f};
    const float* wcol = Wout + (size_t)(col0 + R) * HH;  // B[k][n] = Wout[n*H + k]
#pragma unroll
    for (int kk = 0; kk < 16; ++kk) {
      v2f bf;
      bf.x = wcol[kk * 4 + koff];
      bf.y = wcol[kk * 4 + koff + 1];
      acc = __builtin_amdgcn_wmma_f32_16x16x4_f32(false, a[kk], false, bf,
                                                  (short)0, acc, false, false);
    }
    // stage tile: (M,N) = (r + 8*half, lane%16)
#pragma unroll
    for (int r = 0; r < 8; ++r) tileS[w][r * 32 + lane] = acc[r];
    __syncthreads();
    // lane handles row R, its half of 16 columns -> online softmax + gumbel-argmax
#pragma unroll
    for (int cc = 0; cc < 8; ++cc) {
      const int cl = half * 8 + cc;
      const int col = col0 + cl;
      float l = tileS[w][rbase + cl] + bout[col];
      float nm = fmaxf(m, l);
      float e1 = __expf(m - nm);
      float e2 = __expf(l - nm);
      s = s * e1 + e2;
      tsum = tsum * e1 + l * e2;
      m = nm;
      float gv = l + grow[col];
      if (gv > bv) { bv = gv; bi = col; }
    }
    __syncthreads();
  }

  // cross-lane / cross-wave merge via LDS
  sm_[tid] = m; ss_[tid] = s; st_[tid] = tsum; sbv_[tid] = bv; sbi_[tid] = bi;
  __syncthreads();
  if (tid < 16) {
    float M = -3.0e38f, S = 0.0f, T = 0.0f, BV = -3.0e38f;
    int BI = 0x7fffffff;
    for (int q = 0; q < 8; ++q) {
      const int src = (q >> 1) * 32 + tid + ((q & 1) << 4);
      float m2 = sm_[src], s2 = ss_[src], t2 = st_[src];
      float nm = fmaxf(M, m2);
      float e1 = __expf(M - nm), e2 = __expf(m2 - nm);
      S = S * e1 + s2 * e2;
      T = T * e1 + t2 * e2;
      M = nm;
      float bv2 = sbv_[src]; int bi2 = sbi_[src];
      if (bv2 > BV || (bv2 == BV && bi2 < BI)) { BV = bv2; BI = bi2; }
    }
    const size_t gi = (size_t)(row0 + tid) * NSEG + seg;
    pm[gi] = M; ps[gi] = S; pt[gi] = T; pbv[gi] = BV; pbi[gi] = BI;
  }
}

// -------------------- per step: combine segments, pick token, scatter one-hot --------------------
__global__ void k_select(const float* __restrict__ pm, const float* __restrict__ ps,
                         const float* __restrict__ pt, const float* __restrict__ pbv,
                         const int* __restrict__ pbi, float* __restrict__ ent,
                         int* __restrict__ tok, int* __restrict__ seq,
                         float* __restrict__ out, int step) {
  const int b = blockIdx.x * blockDim.x + threadIdx.x;
  if (b >= BB) return;
  float M = -3.0e38f, S = 0.0f, T = 0.0f, BV = -3.0e38f;
  int BI = 0x7fffffff;
  for (int q = 0; q < NSEG; ++q) {
    const size_t gi = (size_t)b * NSEG + q;
    float m2 = pm[gi], s2 = ps[gi], t2 = pt[gi];
    float nm = fmaxf(M, m2);
    float e1 = __expf(M - nm), e2 = __expf(m2 - nm);
    S = S * e1 + s2 * e2;
    T = T * e1 + t2 * e2;
    M = nm;
    float bv2 = pbv[gi]; int bi2 = pbi[gi];
    if (bv2 > BV || (bv2 == BV && bi2 < BI)) { BV = bv2; BI = bi2; }
  }
  const float logZ = M + __logf(S);
  ent[b] += logZ - T / S;                       // -sum(p*logp)
  tok[b] = BI;
  out[OUT_TOK + ((size_t)b * (LL + 1) + step + 1) * VV + BI] = 1.0f;
  if (BI == EOS && seq[b] == LL + 1) seq[b] = step + 2;
}

// -------------------- finalize --------------------
__global__ void k_final(const float* __restrict__ ent, const int* __restrict__ seq,
                        float* __restrict__ out) {
  __shared__ float red[256];
  const int t = threadIdx.x;
  float s = 0.0f;
  for (int i = t; i < BB; i += 256) {
    s += ent[i];
    out[OUT_SEQ + i] = (float)seq[i];
  }
  red[t] = s;
  __syncthreads();
  for (int off = 128; off > 0; off >>= 1) {
    if (t < off) red[t] += red[t + off];
    __syncthreads();
  }
  if (t == 0) out[OUT_ENT] = red[0] / (float)(BB * LL);
}

extern "C" void kernel_launch(void* const* d_in, const int* in_sizes, int n_in,
                              void* d_out, int out_size, void* d_ws, size_t ws_size,
                              hipStream_t stream) {
  (void)in_sizes; (void)n_in; (void)ws_size;
  const float* hin  = (const float*)d_in[0];
  const float* emb  = (const float*)d_in[1];
  const float* Wih  = (const float*)d_in[2];
  const float* Whh  = (const float*)d_in[3];
  const float* bih  = (const float*)d_in[4];
  const float* bhh  = (const float*)d_in[5];
  const float* Wout = (const float*)d_in[6];
  const float* bout = (const float*)d_in[7];
  const float* gum  = (const float*)d_in[8];
  float* out = (float*)d_out;

  float* ws = (float*)d_ws;
  float* h   = ws + WS_H;
  float* c   = ws + WS_C;
  float* ent = ws + WS_ENT;
  float* pm  = ws + WS_PM;
  float* ps  = ws + WS_PS;
  float* pt  = ws + WS_PT;
  float* pbv = ws + WS_PBV;
  int*   tok = (int*)(ws + WS_TOK);
  int*   seq = (int*)(ws + WS_SEQ);
  int*   pbi = (int*)(ws + WS_PBI);

  // zero entire output (one-hots + pads); graph-capturable memset node
  hipMemsetAsync(d_out, 0, (size_t)out_size * sizeof(float), stream);

  k_init<<<(BB * HH + 255) / 256, 256, 0, stream>>>(hin, h, c, ent, tok, seq, out);

  for (int step = 0; step < LL; ++step) {
    k_lstm<<<BB, 256, 0, stream>>>(emb, Wih, Whh, bih, bhh, tok, h, c, out + OUT_EMB, step);
    k_logits<<<dim3(BB / 16, NSEG), 128, 0, stream>>>(h, Wout, bout, gum, step,
                                                      pm, ps, pt, pbv, pbi);
    k_select<<<(BB + 255) / 256, 256, 0, stream>>>(pm, ps, pt, pbv, pbi, ent, tok, seq,
                                                   out, step);
  }

  k_final<<<1, 256, 0, stream>>>(ent, seq, out);
}